// GCN_GRU_81131932221696
// MI455X (gfx1250) — compile-verified
//
#include <hip/hip_runtime.h>
#include <math.h>

// ---------------------------------------------------------------------------
// Problem constants (from reference)
// ---------------------------------------------------------------------------
#define TT      12
#define NN      30000
#define EE      300000
#define INC     128     // input channels
#define HID     256     // hidden of GCN layer 1
#define OUTC    128     // GCN layer 2 output / GRU hidden
#define G3      384     // 3 * OUTC (GRU gates)

typedef __attribute__((ext_vector_type(2))) float v2f;
typedef __attribute__((ext_vector_type(8))) float v8f;

// ---------------------------------------------------------------------------
// Small utility kernels
// ---------------------------------------------------------------------------
__global__ void zero_f32(float* p, int n) {
    int i = blockIdx.x * blockDim.x + threadIdx.x;
    if (i < n) p[i] = 0.0f;
}

__global__ void zero_i32(int* p, int n) {
    int i = blockIdx.x * blockDim.x + threadIdx.x;
    if (i < n) p[i] = 0;
}

// x[n, 0:32]=emb0[cat0], [32:64]=emb1[cat1], [64:96]=emb2[cat2], [96:128]=num
__global__ void gather_features(const int* __restrict__ cat,
                                const float* __restrict__ numx,
                                const float* __restrict__ e0,
                                const float* __restrict__ e1,
                                const float* __restrict__ e2,
                                float* __restrict__ x) {
    int n = blockIdx.x;
    int c = threadIdx.x;   // 0..127
    float v;
    if (c < 32)       v = e0[(size_t)cat[n * 3 + 0] * 32 + c];
    else if (c < 64)  v = e1[(size_t)cat[n * 3 + 1] * 32 + (c - 32)];
    else if (c < 96)  v = e2[(size_t)cat[n * 3 + 2] * 32 + (c - 64)];
    else              v = numx[(size_t)n * 32 + (c - 96)];
    x[(size_t)n * INC + c] = v;
}

// integer in-degree count over edge destinations
__global__ void count_deg(const int* __restrict__ dst, int* __restrict__ deg, int e) {
    int i = blockIdx.x * blockDim.x + threadIdx.x;
    if (i < e)
        __hip_atomic_fetch_add(&deg[dst[i]], 1,
                               __ATOMIC_RELAXED, __HIP_MEMORY_SCOPE_AGENT);
}

// dinv = rsqrt(in-degree + 1 self loop)
__global__ void deg_to_dinv(const int* __restrict__ deg, float* __restrict__ dinv, int n) {
    int i = blockIdx.x * blockDim.x + threadIdx.x;
    if (i < n) dinv[i] = rsqrtf((float)(deg[i] + 1));
}

// single-workgroup exclusive prefix scan of deg[0..NN) -> offs[0..NN]
// (1024 threads, Hillis-Steele in LDS, running carry)
__global__ void __launch_bounds__(1024)
exclusive_scan(const int* __restrict__ deg, int* __restrict__ offs) {
    __shared__ int tmp[1024];
    __shared__ int s_carry;
    const int tid = threadIdx.x;
    if (tid == 0) s_carry = 0;
    for (int base = 0; base < NN; base += 1024) {
        int i = base + tid;
        int v = (i < NN) ? deg[i] : 0;
        tmp[tid] = v;
        __syncthreads();                       // also publishes s_carry update
        for (int off = 1; off < 1024; off <<= 1) {
            int add = (tid >= off) ? tmp[tid - off] : 0;
            __syncthreads();
            tmp[tid] += add;
            __syncthreads();
        }
        int c = s_carry;
        if (i < NN) offs[i] = c + tmp[tid] - v;  // exclusive
        __syncthreads();
        if (tid == 0) s_carry = c + tmp[1023];
    }
    __syncthreads();
    if (tid == 0) offs[NN] = s_carry;
}

// place each edge's src into its destination's CSR bucket (1 int atomic/edge)
__global__ void fill_csr(const int* __restrict__ src, const int* __restrict__ dst,
                         const int* __restrict__ offs, int* __restrict__ cursor,
                         int* __restrict__ csr_src, int e) {
    int i = blockIdx.x * blockDim.x + threadIdx.x;
    if (i < e) {
        int d = dst[i];
        int pos = __hip_atomic_fetch_add(&cursor[d], 1,
                                         __ATOMIC_RELAXED, __HIP_MEMORY_SCOPE_AGENT);
        csr_src[offs[d] + pos] = src[i];
    }
}

// gather-style GCN aggregation: one block per destination node, C lanes.
// out[n,c] = bias[c] + dinv[n]^2 * h[n,c] + sum_in dinv[s]*dinv[n]*h[s,c]
template <int C>
__global__ void __launch_bounds__(C)
gcn_aggregate(const int* __restrict__ offs, const int* __restrict__ csr_src,
              const float* __restrict__ dinv, const float* __restrict__ h,
              const float* __restrict__ bias, float* __restrict__ out) {
    int n = blockIdx.x;
    int c = threadIdx.x;
    float dn = dinv[n];
    float acc = bias[c] + h[(size_t)n * C + c] * dn * dn;   // self loop + bias
    int beg = offs[n], end = offs[n + 1];
    for (int i = beg; i < end; ++i) {
        int s = csr_src[i];
        acc += h[(size_t)s * C + c] * (dinv[s] * dn);
    }
    out[(size_t)n * C + c] = acc;
}

// GRU gate pointwise: h = (1-z)*n + z*h
__global__ void gru_pointwise(const float* __restrict__ gi,
                              const float* __restrict__ gh,
                              float* __restrict__ h, int total) {
    int i = blockIdx.x * blockDim.x + threadIdx.x;
    if (i >= total) return;
    int n = i >> 7;          // / 128
    int c = i & 127;
    size_t base = (size_t)n * G3;
    float ir = gi[base + c],        hr = gh[base + c];
    float iz = gi[base + 128 + c],  hz = gh[base + 128 + c];
    float in_ = gi[base + 256 + c], hn = gh[base + 256 + c];
    float r = 1.0f / (1.0f + __expf(-(ir + hr)));
    float z = 1.0f / (1.0f + __expf(-(iz + hz)));
    float nn = tanhf(in_ + r * hn);
    float hp = h[i];
    h[i] = (1.0f - z) * nn + z * hp;
}

// ---------------------------------------------------------------------------
// FP32 WMMA GEMM:  C[N x OUTW] = op(A)[N x K] * B[K x OUTW] (+ bias)
//   one wave per 16x64 output strip (4 adjacent 16x16 N-tiles): the A fragment
//   is loaded once per k-step (one b64 per lane) and feeds 4 WMMAs.
//   V_WMMA_F32_16X16X4_F32, K step 4. Full fp32 precision.
//   TRANSB=false: B row-major [K][OUTW]; TRANSB=true: B is W[OUTW][K] (B=W^T)
//   RELU_A: apply relu to A elements on load (fuses GCN relu into layer 2)
// A-matrix VGPR layout (ISA 7.12.2, 32-bit A 16x4): lanes 0-15 -> K={k,k+1},
// lanes 16-31 -> K={k+2,k+3}; C/D: VGPR i holds row (i + 8*(lane>=16)).
// ---------------------------------------------------------------------------
template <int K, int OUTW, bool TRANSB, bool RELU_A>
__global__ void __launch_bounds__(32)
wmma_gemm_f32(const float* __restrict__ A,
              const float* __restrict__ B,
              const float* __restrict__ bias,
              float* __restrict__ C) {
    const int lane = threadIdx.x & 31;
    const int half = lane >> 4;        // 0: lanes 0-15, 1: lanes 16-31
    const int l    = lane & 15;
    const int m0   = blockIdx.x * 16;
    const int n0   = blockIdx.y * 64;  // 4 N-tiles per wave

    v8f acc[4] = {v8f{}, v8f{}, v8f{}, v8f{}};
    const float* arow = A + (size_t)(m0 + l) * K;

#pragma unroll 8
    for (int k = 0; k < K; k += 4) {
        const int ak = k + half * 2;

        // A fragment: two consecutive floats -> one b64 load, shared by 4 WMMAs
        v2f av = *reinterpret_cast<const v2f*>(arow + ak);
        if (RELU_A) { av.x = fmaxf(av.x, 0.0f); av.y = fmaxf(av.y, 0.0f); }

#pragma unroll
        for (int j = 0; j < 4; ++j) {
            const int col = n0 + j * 16 + l;
            v2f bv;
            if (TRANSB) {   // B[k][col] = W[col*K + k] : consecutive -> b64
                bv = *reinterpret_cast<const v2f*>(B + (size_t)col * K + ak);
            } else {        // row-major B
                bv.x = B[(size_t)ak * OUTW + col];
                bv.y = B[(size_t)(ak + 1) * OUTW + col];
            }
            acc[j] = __builtin_amdgcn_wmma_f32_16x16x4_f32(
                /*neg_a=*/false, av, /*neg_b=*/false, bv,
                /*c_mod=*/(short)0, acc[j], /*reuse_a=*/false, /*reuse_b=*/false);
        }
    }

#pragma unroll
    for (int j = 0; j < 4; ++j) {
        const int col = n0 + j * 16 + l;
        const float bb = bias ? bias[col] : 0.0f;
#pragma unroll
        for (int i = 0; i < 8; ++i) {
            C[(size_t)(m0 + half * 8 + i) * OUTW + col] = acc[j][i] + bb;
        }
    }
}

// ---------------------------------------------------------------------------
// Host driver
// ---------------------------------------------------------------------------
extern "C" void kernel_launch(void* const* d_in, const int* in_sizes, int n_in,
                              void* d_out, int out_size, void* d_ws, size_t ws_size,
                              hipStream_t stream) {
    (void)in_sizes; (void)n_in; (void)out_size; (void)ws_size;

    const int*   cat_x = (const int*)  d_in[0];   // [T, N, 3]
    const float* num_x = (const float*)d_in[1];   // [T, N, 32]
    const int*   edges = (const int*)  d_in[2];   // [T, 2, E]
    const float* emb0  = (const float*)d_in[3];   // [1000, 32]
    const float* emb1  = (const float*)d_in[4];
    const float* emb2  = (const float*)d_in[5];
    const float* W1    = (const float*)d_in[6];   // [T, 128, 256]
    const float* b1    = (const float*)d_in[7];   // [T, 256]
    const float* W2    = (const float*)d_in[8];   // [T, 256, 128]
    const float* b2    = (const float*)d_in[9];   // [T, 128]
    const float* w_ih  = (const float*)d_in[10];  // [384, 128]
    const float* w_hh  = (const float*)d_in[11];  // [384, 128]
    const float* b_ih  = (const float*)d_in[12];  // [384]
    const float* b_hh  = (const float*)d_in[13];  // [384]

    float* h = (float*)d_out;                     // GRU hidden, [N, 128] — output

    // workspace carve-up: floats first, then ints
    float* wf  = (float*)d_ws;
    float* x    = wf; wf += (size_t)NN * INC;     // [N,128] gathered features
    float* h1   = wf; wf += (size_t)NN * HID;     // [N,256] x @ W1
    float* g1   = wf; wf += (size_t)NN * HID;     // [N,256] layer-1 out (pre-relu)
    float* h2   = wf; wf += (size_t)NN * OUTC;    // [N,128] relu(g1) @ W2
    float* g2   = wf; wf += (size_t)NN * OUTC;    // [N,128] layer-2 out
    float* dinv = wf; wf += (size_t)NN;           // [N] 1/sqrt(deg)
    float* gi   = wf; wf += (size_t)NN * G3;      // [N,384]
    float* gh   = wf; wf += (size_t)NN * G3;      // [N,384]
    int*   wi   = (int*)wf;
    int*   deg     = wi; wi += NN;                // [N]  in-degree
    int*   cursor  = wi; wi += NN;                // [N]  CSR fill cursor
    int*   offs    = wi; wi += NN + 1;            // [N+1] CSR row offsets
    int*   csr_src = wi; wi += EE;                // [E]  src node per CSR slot

    const int TPB = 256;

    // h0 = 0
    zero_f32<<<(NN * OUTC + TPB - 1) / TPB, TPB, 0, stream>>>(h, NN * OUTC);

    const dim3 grid1(NN / 16, HID / 64);   // 1875 x 4
    const dim3 grid2(NN / 16, OUTC / 64);  // 1875 x 2
    const dim3 gridG(NN / 16, G3 / 64);    // 1875 x 6

    for (int t = 0; t < TT; ++t) {
        const int*   cat_t = cat_x + (size_t)t * NN * 3;
        const float* num_t = num_x + (size_t)t * NN * 32;
        const int*   src   = edges + (size_t)t * 2 * EE;
        const int*   dst   = src + EE;

        // features
        gather_features<<<NN, INC, 0, stream>>>(cat_t, num_t, emb0, emb1, emb2, x);

        // ---- per-t CSR build + symmetric degree norm ----
        zero_i32<<<(NN + TPB - 1) / TPB, TPB, 0, stream>>>(deg, NN);
        zero_i32<<<(NN + TPB - 1) / TPB, TPB, 0, stream>>>(cursor, NN);
        count_deg<<<(EE + TPB - 1) / TPB, TPB, 0, stream>>>(dst, deg, EE);
        deg_to_dinv<<<(NN + TPB - 1) / TPB, TPB, 0, stream>>>(deg, dinv, NN);
        exclusive_scan<<<1, 1024, 0, stream>>>(deg, offs);
        fill_csr<<<(EE + TPB - 1) / TPB, TPB, 0, stream>>>(src, dst, offs, cursor,
                                                           csr_src, EE);

        // ---- GCN layer 1: h1 = x @ W1[t]; g1 = aggregate(h1) + b1[t] ----
        wmma_gemm_f32<INC, HID, false, false><<<grid1, 32, 0, stream>>>(
            x, W1 + (size_t)t * INC * HID, nullptr, h1);
        gcn_aggregate<HID><<<NN, HID, 0, stream>>>(offs, csr_src, dinv, h1,
                                                   b1 + (size_t)t * HID, g1);

        // ---- GCN layer 2: h2 = relu(g1) @ W2[t]; g2 = aggregate(h2) + b2[t] ----
        wmma_gemm_f32<HID, OUTC, false, true><<<grid2, 32, 0, stream>>>(
            g1, W2 + (size_t)t * HID * OUTC, nullptr, h2);
        gcn_aggregate<OUTC><<<NN, OUTC, 0, stream>>>(offs, csr_src, dinv, h2,
                                                     b2 + (size_t)t * OUTC, g2);

        // ---- GRU step: gi = g2 @ w_ih^T + b_ih ; gh = h @ w_hh^T + b_hh ----
        wmma_gemm_f32<OUTC, G3, true, false><<<gridG, 32, 0, stream>>>(
            g2, w_ih, b_ih, gi);
        wmma_gemm_f32<OUTC, G3, true, false><<<gridG, 32, 0, stream>>>(
            h, w_hh, b_hh, gh);
        gru_pointwise<<<(NN * OUTC + TPB - 1) / TPB, TPB, 0, stream>>>(gi, gh, h, NN * OUTC);
    }
}